// pyramid_step_model_55413668053134
// MI455X (gfx1250) — compile-verified
//
#include <hip/hip_runtime.h>
#include <stdint.h>

// ---------------- constants (match reference shapes) ----------------
constexpr int B_    = 8;
constexpr int N_    = 131072;     // points per batch (2^17)
constexpr int NF_   = 8;          // features per point
constexpr int NQ    = 64;         // n1 = n2 = 64
constexpr int M1    = 2048;       // elements per level-1 bucket
constexpr int M_    = 32;         // elements per level-2 bucket
constexpr int NP    = 66;         // padded grid 66x66
constexpr int HI    = 132;        // upsampled 132x132
constexpr int NREG  = 64;
constexpr int CH    = M_ * NF_;   // 256 channels (m*8+f)
constexpr int MP    = 144;        // padded M for GEMM (132 -> 144)
constexpr int KP    = 68;         // padded K for GEMM (66 -> 68)
constexpr int N1SZ  = CH * KP;    // pass-1 N  = 17408  (= 64*272)
constexpr int N2SZ  = CH * MP;    // pass-2 N  = 36864  (= 64*576)

typedef __attribute__((ext_vector_type(2))) float        v2f;
typedef __attribute__((ext_vector_type(8))) float        v8f;
typedef __attribute__((ext_vector_type(4))) unsigned int v4u;
typedef __attribute__((ext_vector_type(8))) int          v8i;
typedef __attribute__((ext_vector_type(4))) int          v4i;

// ---------------- stage 0: init sort arrays ----------------
__global__ void k_init_kv(const float* __restrict__ coords, float* __restrict__ key,
                          int* __restrict__ val) {
    int t = blockIdx.x * blockDim.x + threadIdx.x;           // over B*N
    if (t >= B_ * N_) return;
    int b = t / N_, i = t % N_;
    key[t] = coords[(long)b * 2 * N_ + i];                   // x coordinate
    val[t] = i;
}

// ---------------- bitonic sort pieces ----------------
__global__ __launch_bounds__(1024) void k_presort(float* __restrict__ key, int* __restrict__ val) {
    __shared__ float sk[2048];
    __shared__ int   sv[2048];
    int chunk = blockIdx.x;
    long base = (long)chunk * 2048;
    int t = threadIdx.x;
    sk[t]        = key[base + t];        sv[t]        = val[base + t];
    sk[t + 1024] = key[base + t + 1024]; sv[t + 1024] = val[base + t + 1024];
    int localBase = (chunk & 63) * 2048;
    for (int k = 2; k <= 2048; k <<= 1) {
        for (int j = k >> 1; j > 0; j >>= 1) {
            __syncthreads();
            int i = ((t & ~(j - 1)) << 1) | (t & (j - 1));
            bool up = (((localBase + i) & k) == 0);
            float a = sk[i], b = sk[i + j];
            bool sw = up ? (a > b) : (a < b);
            if (sw) {
                sk[i] = b; sk[i + j] = a;
                int tv = sv[i]; sv[i] = sv[i + j]; sv[i + j] = tv;
            }
        }
    }
    __syncthreads();
    key[base + t]        = sk[t];        val[base + t]        = sv[t];
    key[base + t + 1024] = sk[t + 1024]; val[base + t + 1024] = sv[t + 1024];
}

__global__ void k_gpass(float* __restrict__ key, int* __restrict__ val, int k, int j) {
    int t = blockIdx.x * blockDim.x + threadIdx.x;           // over B*N/2
    if (t >= B_ * (N_ / 2)) return;
    int seg = t / (N_ / 2), tl = t % (N_ / 2);
    int i = ((tl & ~(j - 1)) << 1) | (tl & (j - 1));
    long bi = (long)seg * N_ + i;
    bool up = ((i & k) == 0);
    float a = key[bi], b = key[bi + j];
    bool sw = up ? (a > b) : (a < b);
    if (sw) {
        key[bi] = b; key[bi + j] = a;
        int tv = val[bi]; val[bi] = val[bi + j]; val[bi + j] = tv;
    }
}

__global__ __launch_bounds__(1024) void k_finish(float* __restrict__ key, int* __restrict__ val, int k) {
    __shared__ float sk[2048];
    __shared__ int   sv[2048];
    int chunk = blockIdx.x;
    long base = (long)chunk * 2048;
    int t = threadIdx.x;
    sk[t]        = key[base + t];        sv[t]        = val[base + t];
    sk[t + 1024] = key[base + t + 1024]; sv[t + 1024] = val[base + t + 1024];
    int localBase = (chunk & 63) * 2048;
    for (int j = 1024; j > 0; j >>= 1) {
        __syncthreads();
        int i = ((t & ~(j - 1)) << 1) | (t & (j - 1));
        bool up = (((localBase + i) & k) == 0);
        float a = sk[i], b = sk[i + j];
        bool sw = up ? (a > b) : (a < b);
        if (sw) {
            sk[i] = b; sk[i + j] = a;
            int tv = sv[i]; sv[i] = sv[i + j]; sv[i + j] = tv;
        }
    }
    __syncthreads();
    key[base + t]        = sk[t];        val[base + t]        = sv[t];
    key[base + t + 1024] = sk[t + 1024]; val[base + t + 1024] = sv[t + 1024];
}

// ---------------- level-2 bucketing ----------------
__global__ __launch_bounds__(1024) void k_level2(const float* __restrict__ coords,
                                                 const int* __restrict__ sval,
                                                 int* __restrict__ idx_tot,
                                                 float* __restrict__ cm0,
                                                 float* __restrict__ cm1) {
    __shared__ float yk[2048];
    __shared__ int   yv[2048];
    __shared__ float xv[2048];
    int blk = blockIdx.x;                   // b*64 + xb
    int b = blk >> 6, xb = blk & 63;
    long base = (long)b * N_ + (long)xb * M1;
    int t = threadIdx.x;
    for (int e = t; e < 2048; e += 1024) {
        int oi = sval[base + e];
        yk[e] = coords[(long)b * 2 * N_ + N_ + oi];
        yv[e] = oi;
    }
    for (int k = 2; k <= 2048; k <<= 1) {
        for (int j = k >> 1; j > 0; j >>= 1) {
            __syncthreads();
            int i = ((t & ~(j - 1)) << 1) | (t & (j - 1));
            bool up = ((i & k) == 0);
            float a = yk[i], bb = yk[i + j];
            bool sw = up ? (a > bb) : (a < bb);
            if (sw) {
                yk[i] = bb; yk[i + j] = a;
                int tv = yv[i]; yv[i] = yv[i + j]; yv[i + j] = tv;
            }
        }
    }
    __syncthreads();
    for (int e = t; e < 2048; e += 1024)
        xv[e] = coords[(long)b * 2 * N_ + yv[e]];
    __syncthreads();
    if (t < 64) {
        int g = t, o = g * 32;
        float medy = yk[o + 15];
        for (int a2 = 1; a2 < 32; ++a2) {
            int kv = yv[o + a2]; int p = a2 - 1;
            while (p >= 0 && yv[o + p] > kv) { yv[o + p + 1] = yv[o + p]; --p; }
            yv[o + p + 1] = kv;
        }
        for (int a2 = 1; a2 < 32; ++a2) {
            float kv = xv[o + a2]; int p = a2 - 1;
            while (p >= 0 && xv[o + p] > kv) { xv[o + p + 1] = xv[o + p]; --p; }
            xv[o + p + 1] = kv;
        }
        int cmi = (b * NQ + xb) * NQ + g;
        cm0[cmi] = xv[o + 15];
        cm1[cmi] = medy;
        long ib = ((long)(b * NQ + xb) * NQ + g) * M_;
        for (int m = 0; m < 32; ++m) idx_tot[ib + m] = yv[o + m];
    }
}

// ---------------- scaled/padded coords [B,2,66,66] ----------------
__global__ void k_scale(const float* __restrict__ cm0, const float* __restrict__ cm1,
                        float* __restrict__ sc) {
    __shared__ float mn0[64], mx0[64], mn1[64], mx1[64];
    int b = blockIdx.x, t = threadIdx.x;
    if (t < 64) {
        int j = t; float mn = 1e30f, mx = -1e30f;
        for (int i = 0; i < 64; ++i) {
            float v = cm0[(b * NQ + i) * NQ + j];
            mn = fminf(mn, v); mx = fmaxf(mx, v);
        }
        mn0[j] = mn - 0.5f; mx0[j] = mx + 0.5f;
    } else if (t < 128) {
        int i = t - 64; float mn = 1e30f, mx = -1e30f;
        for (int j = 0; j < 64; ++j) {
            float v = cm1[(b * NQ + i) * NQ + j];
            mn = fminf(mn, v); mx = fmaxf(mx, v);
        }
        mn1[i] = mn - 0.5f; mx1[i] = mx + 0.5f;
    }
    __syncthreads();
    for (int e = t; e < NP * NP; e += blockDim.x) {
        int I = e / NP, J = e % NP;
        int jj = min(max(J - 1, 0), 63);
        int ii = min(max(I - 1, 0), 63);
        float v0 = (I == 0) ? mn0[jj] : ((I == 65) ? mx0[jj] : cm0[(b * NQ + (I - 1)) * NQ + jj]);
        float v1 = (J == 0) ? mn1[ii] : ((J == 65) ? mx1[ii] : cm1[(b * NQ + ii) * NQ + (J - 1)]);
        sc[((long)(b * 2 + 0) * NP + I) * NP + J] = v0;
        sc[((long)(b * 2 + 1) * NP + I) * NP + J] = v1;
    }
}

// ---------------- bilinear 66->132 on coords ----------------
__global__ void k_bilin(const float* __restrict__ sc, float* __restrict__ ci) {
    int t = blockIdx.x * blockDim.x + threadIdx.x;
    if (t >= B_ * 2 * HI * HI) return;
    int wo = t % HI, ho = (t / HI) % HI, c = (t / (HI * HI)) % 2, b = t / (2 * HI * HI);
    const float s = 65.0f / 131.0f;
    float ph = ho * s, pw = wo * s;
    int h0 = (int)floorf(ph), w0 = (int)floorf(pw);
    h0 = min(max(h0, 0), 65); w0 = min(max(w0, 0), 65);
    int h1 = min(h0 + 1, 65), w1 = min(w0 + 1, 65);
    float fh = ph - h0, fw = pw - w0;
    const float* p = sc + (long)(b * 2 + c) * NP * NP;
    float v = (1 - fh) * ((1 - fw) * p[h0 * NP + w0] + fw * p[h0 * NP + w1]) +
              fh       * ((1 - fw) * p[h1 * NP + w0] + fw * p[h1 * NP + w1]);
    ci[t] = v;
}

// ---------------- argmin selections ----------------
__global__ void k_ind0(const float* __restrict__ ci, int* __restrict__ ind0) {
    int t = blockIdx.x * blockDim.x + threadIdx.x;
    if (t >= B_ * HI * NREG) return;
    int n = t % NREG, w = (t / NREG) % HI, b = t / (NREG * HI);
    float g = (float)n * (1.0f / 63.0f);
    const float* c0 = ci + (long)(b * 2 + 0) * HI * HI;
    float best = 1e30f; int bi = 0;
    for (int h = 0; h < HI; ++h) {
        float d = fabsf(c0[h * HI + w] - g);
        if (d < best) { best = d; bi = h; }
    }
    ind0[t] = bi;
}

__global__ void k_sel(const float* __restrict__ ci, const int* __restrict__ ind0,
                      int* __restrict__ Hsel, int* __restrict__ Wsel) {
    int t = blockIdx.x * blockDim.x + threadIdx.x;
    if (t >= B_ * NREG * NREG) return;
    int np = t % NREG, n = (t / NREG) % NREG, b = t / (NREG * NREG);
    float g = (float)np * (1.0f / 63.0f);
    const float* c1 = ci + (long)(b * 2 + 1) * HI * HI;
    float best = 1e30f; int bw = 0;
    for (int w = 0; w < HI; ++w) {
        int h = ind0[(b * HI + w) * NREG + n];
        float d = fabsf(c1[h * HI + w] - g);
        if (d < best) { best = d; bw = w; }
    }
    Hsel[t] = ind0[(b * HI + bw) * NREG + n];
    Wsel[t] = bw;
}

// ---------------- bicubic weight matrix W [144 x 68] ----------------
__device__ inline float cubic_w(float x) {
    const float a = -0.75f;
    float at = fabsf(x);
    if (at <= 1.0f) return ((a + 2.0f) * at - (a + 3.0f)) * at * at + 1.0f;
    if (at < 2.0f)  return (((at - 5.0f) * at + 8.0f) * at - 4.0f) * a;
    return 0.0f;
}

__global__ void k_winit(float* __restrict__ Wm) {
    int t = blockIdx.x * blockDim.x + threadIdx.x;
    if (t >= MP * KP) return;
    int m = t / KP, k = t % KP;
    float w = 0.0f;
    if (m < HI && k < NP) {
        float pos = (float)m * (65.0f / 131.0f);
        int i0 = (int)floorf(pos);
        float frac = pos - (float)i0;
        for (int tp = -1; tp <= 2; ++tp) {
            int idx = min(max(i0 + tp, 0), 65);
            if (idx == k) w += cubic_w(frac - (float)tp);
        }
    }
    Wm[t] = w;
}

// ---------------- build padded field as pass-1 B matrix, COLUMN-MAJOR [N][K] ----------------
// B1t[n*KP + k] with n = ch*68 + J, k = I.  I is innermost -> coalesced writes.
__global__ void k_buildS(const float* __restrict__ x, const int* __restrict__ idx_tot,
                         float* __restrict__ B1t, int b) {
    int t = blockIdx.x * blockDim.x + threadIdx.x;
    if (t >= CH * KP * KP) return;
    int I = t % KP, J = (t / KP) % KP, ch = t / (KP * KP);
    float v = 0.0f;
    if (I < NP && J < NP) {
        int ii = min(max(I - 1, 0), 63);
        int jj = min(max(J - 1, 0), 63);
        int m = ch >> 3, f = ch & 7;
        int oi = idx_tot[((long)(b * NQ + ii) * NQ + jj) * M_ + m];
        v = x[((long)b * N_ + oi) * NF_ + f];
    }
    B1t[(long)(ch * KP + J) * KP + I] = v;
}

// ---------------- TDM: 2D tile (float) global -> LDS ----------------
// Descriptor per CDNA5 ISA 8.3/8.4.  All inputs wave-uniform.
__device__ inline void tdm_load_2d(uint32_t lds_off, const float* gptr,
                                   uint32_t tile_w, uint32_t tile_h,
                                   uint32_t row_stride, uint32_t tensor_w,
                                   uint32_t tensor_h) {
#if __has_builtin(__builtin_amdgcn_tensor_load_to_lds)
    uint64_t ga = (uint64_t)(uintptr_t)gptr;
    v4u g0;
    g0[0] = 1u;                                                // count=1, user descriptor
    g0[1] = lds_off;                                           // lds_addr (bytes)
    g0[2] = (uint32_t)(ga & 0xFFFFFFFFu);                      // global_addr[31:0]
    g0[3] = (uint32_t)((ga >> 32) & 0x01FFFFFFu) | (2u << 30); // addr[56:32] | type=2
    uint64_t s0 = (uint64_t)row_stride;                        // tensor_dim0_stride (elems)
    v8i g1;
    g1[0] = (int)(2u << 16);                                   // data_size = 4 bytes
    g1[1] = (int)((tensor_w & 0xFFFFu) << 16);                 // tensor_dim0[15:0]
    g1[2] = (int)(((tensor_w >> 16) & 0xFFFFu) | ((tensor_h & 0xFFFFu) << 16));
    g1[3] = (int)(((tensor_h >> 16) & 0xFFFFu) | ((tile_w & 0xFFFFu) << 16)); // tile_dim0
    g1[4] = (int)(tile_h & 0xFFFFu);                           // tile_dim1 (tile_dim2=0)
    g1[5] = (int)(uint32_t)(s0 & 0xFFFFFFFFu);                 // stride0[31:0]
    g1[6] = (int)(uint32_t)((s0 >> 32) & 0xFFFFu);             // stride0[47:32], stride1=0
    g1[7] = 0;
    v4i z4 = {0, 0, 0, 0};
#if __has_include(<hip/amd_detail/amd_gfx1250_TDM.h>)
    v8i z8 = {0, 0, 0, 0, 0, 0, 0, 0};
    __builtin_amdgcn_tensor_load_to_lds(g0, g1, z4, z4, z8, 0);   // clang-23 6-arg form
#else
    __builtin_amdgcn_tensor_load_to_lds(g0, g1, z4, z4, 0);       // ROCm 7.2 5-arg form
#endif
#else
    (void)lds_off; (void)gptr; (void)tile_w; (void)tile_h;
    (void)row_stride; (void)tensor_w; (void)tensor_h;
#endif
}

// ---------------- WMMA GEMM: D[144 x N] = W[144x68] * B[N][K]^T ----------------
// B is stored column-major ([N rows][K cols], row-major), so each WMMA B fragment
// is a single aligned b64 LDS load (rows k0,k0+1 adjacent).  One wave computes a
// 16(M) x 64(N) strip; A (16x68) and B (64x68) tiles staged into LDS by the TDM.
// mode 0: n -> (ch=n/68,  j =n%68 ); store T2t[(ch*144 + m)*68 + j]  ([N][K] for pass 2)
// mode 1: n -> (ch=n/144, h'=n%144); store si[ch][h'][m] when inside 132x132
__global__ __launch_bounds__(32) void k_wmma_gemm(const float* __restrict__ Wm,
                                                  const float* __restrict__ Bt,
                                                  float* __restrict__ dst, int mode) {
    __shared__ float sA[16 * KP];    //  4352 B
    __shared__ float sB[64 * KP];    // 17408 B  ([n][k])
    int lane = threadIdx.x;
    int tN = blockIdx.x, tM = blockIdx.y;

    __builtin_prefetch((const void*)&Bt[(long)tN * 64 * KP], 0, 1);

#if __has_builtin(__builtin_amdgcn_tensor_load_to_lds)
    tdm_load_2d((uint32_t)(uintptr_t)sA, Wm + (size_t)tM * 16 * KP,
                KP, 16, KP, KP, MP);
    tdm_load_2d((uint32_t)(uintptr_t)sB, Bt + (size_t)tN * 64 * KP,
                KP, 64, KP, KP, 64);
    __builtin_amdgcn_s_wait_tensorcnt(0);
    asm volatile("" ::: "memory");
#else
    for (int idx = lane; idx < 16 * KP; idx += 32)
        sA[idx] = Wm[(size_t)tM * 16 * KP + idx];
    for (int idx = lane; idx < 64 * KP; idx += 32)
        sB[idx] = Bt[(long)tN * 64 * KP + idx];
#endif
    __syncthreads();

    int koff = (lane >> 4) << 1;                 // lanes 0-15 -> K{0,1}; 16-31 -> K{2,3}
    int arow = (lane & 15) * KP;
    int brow0 = (lane & 15) * KP;                // N-tile 0 row for this lane
    v8f acc0 = {}, acc1 = {}, acc2 = {}, acc3 = {};
    for (int kt = 0; kt < KP / 4; ++kt) {
        int k0 = kt * 4 + koff;
        v2f a, b0, b1, b2, b3;
        a.x  = sA[arow + k0];              a.y  = sA[arow + k0 + 1];
        b0.x = sB[brow0 + k0];             b0.y = sB[brow0 + k0 + 1];
        b1.x = sB[brow0 + 16 * KP + k0];   b1.y = sB[brow0 + 16 * KP + k0 + 1];
        b2.x = sB[brow0 + 32 * KP + k0];   b2.y = sB[brow0 + 32 * KP + k0 + 1];
        b3.x = sB[brow0 + 48 * KP + k0];   b3.y = sB[brow0 + 48 * KP + k0 + 1];
        acc0 = __builtin_amdgcn_wmma_f32_16x16x4_f32(false, a, false, b0, (short)0, acc0, false, false);
        acc1 = __builtin_amdgcn_wmma_f32_16x16x4_f32(false, a, false, b1, (short)0, acc1, false, false);
        acc2 = __builtin_amdgcn_wmma_f32_16x16x4_f32(false, a, false, b2, (short)0, acc2, false, false);
        acc3 = __builtin_amdgcn_wmma_f32_16x16x4_f32(false, a, false, b3, (short)0, acc3, false, false);
    }

    // direct store from documented C layout: VGPR r <-> M = r + 8*(lane>=16), lane&15 <-> N
    int mbase = tM * 16 + ((lane >> 4) << 3);
    auto store_acc = [&](const v8f& acc, int nt) {
        int gn = tN * 64 + nt * 16 + (lane & 15);
#pragma unroll
        for (int r = 0; r < 8; ++r) {
            int gm = mbase + r;
            float v = acc[r];
            if (mode == 0) {
                int ch = gn / KP, j = gn % KP;
                dst[(long)(ch * MP + gm) * KP + j] = v;
            } else {
                int ch = gn / MP, hp = gn % MP;
                if (gm < HI && hp < HI)
                    dst[(long)ch * (HI * HI) + hp * HI + gm] = v;
            }
        }
    };
    store_acc(acc0, 0);
    store_acc(acc1, 1);
    store_acc(acc2, 2);
    store_acc(acc3, 3);
}

// ---------------- final gather -> output [B,32,8,64,64] ----------------
__global__ void k_gather(const float* __restrict__ si, const int* __restrict__ Hsel,
                         const int* __restrict__ Wsel, float* __restrict__ out, int b) {
    int t = blockIdx.x * blockDim.x + threadIdx.x;
    if (t >= CH * NREG * NREG) return;
    int cc = t % NREG, a = (t / NREG) % NREG, ch = t / (NREG * NREG);
    int sidx = (b * NREG + cc) * NREG + a;
    int h = Hsel[sidx], w = Wsel[sidx];
    int m = ch >> 3, f = ch & 7;
    long oidx = ((((long)(b * M_ + m)) * NF_ + f) * NREG + a) * NREG + cc;
    out[oidx] = si[(long)ch * (HI * HI) + h * HI + w];
}

// ---------------- host launcher ----------------
extern "C" void kernel_launch(void* const* d_in, const int* in_sizes, int n_in,
                              void* d_out, int out_size, void* d_ws, size_t ws_size,
                              hipStream_t stream) {
    (void)in_sizes; (void)n_in; (void)out_size; (void)ws_size;
    const float* x      = (const float*)d_in[0];   // [8,131072,8]
    const float* coords = (const float*)d_in[1];   // [8,2,131072]
    float* out = (float*)d_out;                    // [8,32,8,64,64]

    uint8_t* ws = (uint8_t*)d_ws;
    size_t off = 0;
    auto alloc = [&](size_t bytes) -> void* {
        off = (off + 255) & ~(size_t)255;
        void* p = ws + off; off += bytes; return p;
    };
    float* key    = (float*)alloc((size_t)B_ * N_ * 4);
    int*   val    = (int*)  alloc((size_t)B_ * N_ * 4);
    int*   idxTot = (int*)  alloc((size_t)B_ * NQ * NQ * M_ * 4);
    float* cm0    = (float*)alloc((size_t)B_ * NQ * NQ * 4);
    float* cm1    = (float*)alloc((size_t)B_ * NQ * NQ * 4);
    float* sc     = (float*)alloc((size_t)B_ * 2 * NP * NP * 4);
    float* ci     = (float*)alloc((size_t)B_ * 2 * HI * HI * 4);
    int*   ind0   = (int*)  alloc((size_t)B_ * HI * NREG * 4);
    int*   Hsel   = (int*)  alloc((size_t)B_ * NREG * NREG * 4);
    int*   Wsel   = (int*)  alloc((size_t)B_ * NREG * NREG * 4);
    float* Wm     = (float*)alloc((size_t)MP * KP * 4);
    float* B1t    = (float*)alloc((size_t)N1SZ * KP * 4);    // [N][K], per batch reuse
    float* T2t    = (float*)alloc((size_t)N2SZ * KP * 4);    // [N][K], per batch reuse
    float* si     = (float*)alloc((size_t)CH * HI * HI * 4); // per batch reuse

    // ---- level-1 sort (bitonic) ----
    k_init_kv<<<(B_ * N_ + 255) / 256, 256, 0, stream>>>(coords, key, val);
    k_presort<<<B_ * N_ / 2048, 1024, 0, stream>>>(key, val);
    for (int k = 4096; k <= N_; k <<= 1) {
        for (int j = k >> 1; j >= 2048; j >>= 1)
            k_gpass<<<(B_ * N_ / 2 + 255) / 256, 256, 0, stream>>>(key, val, k, j);
        k_finish<<<B_ * N_ / 2048, 1024, 0, stream>>>(key, val, k);
    }

    // ---- level-2 bucketing + medians ----
    k_level2<<<B_ * NQ, 1024, 0, stream>>>(coords, val, idxTot, cm0, cm1);

    // ---- coords pipeline ----
    k_scale<<<B_, 256, 0, stream>>>(cm0, cm1, sc);
    k_bilin<<<(B_ * 2 * HI * HI + 255) / 256, 256, 0, stream>>>(sc, ci);
    k_ind0<<<(B_ * HI * NREG + 255) / 256, 256, 0, stream>>>(ci, ind0);
    k_sel<<<(B_ * NREG * NREG + 255) / 256, 256, 0, stream>>>(ci, ind0, Hsel, Wsel);

    // ---- bicubic weights ----
    k_winit<<<(MP * KP + 255) / 256, 256, 0, stream>>>(Wm);

    // ---- per-batch field pipeline: build S, two WMMA GEMM passes, gather ----
    for (int b = 0; b < B_; ++b) {
        k_buildS<<<(CH * KP * KP + 255) / 256, 256, 0, stream>>>(x, idxTot, B1t, b);
        k_wmma_gemm<<<dim3(N1SZ / 64, MP / 16), 32, 0, stream>>>(Wm, B1t, T2t, 0);
        k_wmma_gemm<<<dim3(N2SZ / 64, MP / 16), 32, 0, stream>>>(Wm, T2t, si, 1);
        k_gather<<<(CH * NREG * NREG + 255) / 256, 256, 0, stream>>>(si, Hsel, Wsel, out, b);
    }
}